// EmbeddingBlock_53412213293601
// MI455X (gfx1250) — compile-verified
//
#include <hip/hip_runtime.h>

typedef float v2f __attribute__((ext_vector_type(2)));
typedef float v8f __attribute__((ext_vector_type(8)));

#define N_RBF    16
#define ATOM_DIM 128
#define CAT_DIM  272      // 2*128 + 16
#define VOCAB    100

// workspace / LDS layout (floats)
#define TA_OFF   0
#define TB_OFF   (VOCAB * CAT_DIM)          // 27200
#define M2_OFF   (2 * VOCAB * CAT_DIM)      // 54400  (M packed as [8][272][2])
#define PRE_TOT  (M2_OFF + N_RBF * CAT_DIM) // 58752 floats = 235 KB
#define LDS_FLOATS PRE_TOT

#define WAVES 8
#define BLOCK (WAVES * 32)

// ---------------------------------------------------------------------------
// Precompute:  Ta' = emb @ dense_W[:, 0:128].T + bias   -> [100,272]
//              Tb  = emb @ dense_W[:,128:256].T         -> [100,272]
//              M[j,o] = sum_k edge_W[k,j] * dense_W[o,256+k]  -> [16,272],
//                packed pairwise along j for float2 B-fragment loads.
// ---------------------------------------------------------------------------
__global__ void emb_precompute_kernel(const float* __restrict__ edge_W,
                                      const float* __restrict__ emb,
                                      const float* __restrict__ W,
                                      const float* __restrict__ bias,
                                      float* __restrict__ ws) {
  int t = blockIdx.x * blockDim.x + threadIdx.x;
  if (t < TB_OFF) {                       // Ta (bias folded in)
    int v = t / CAT_DIM, o = t % CAT_DIM;
    const float* er = emb + v * ATOM_DIM;
    const float* wr = W + (size_t)o * CAT_DIM;
    float s = bias[o];
#pragma unroll 8
    for (int d = 0; d < ATOM_DIM; ++d) s += er[d] * wr[d];
    ws[TA_OFF + t] = s;
  } else if (t < M2_OFF) {                // Tb
    int tt = t - TB_OFF;
    int v = tt / CAT_DIM, o = tt % CAT_DIM;
    const float* er = emb + v * ATOM_DIM;
    const float* wr = W + (size_t)o * CAT_DIM + ATOM_DIM;
    float s = 0.f;
#pragma unroll 8
    for (int d = 0; d < ATOM_DIM; ++d) s += er[d] * wr[d];
    ws[TB_OFF + tt] = s;
  } else if (t < PRE_TOT) {               // M (folded edge_linear x Wc)
    int tt = t - M2_OFF;
    int j = tt / CAT_DIM, o = tt % CAT_DIM;
    const float* wr = W + (size_t)o * CAT_DIM + 2 * ATOM_DIM;
    float s = 0.f;
#pragma unroll
    for (int k = 0; k < N_RBF; ++k) s += edge_W[k * N_RBF + j] * wr[k];
    // pack: pair p = j>>1, slot = j&1  ->  [p][o][slot]
    ws[M2_OFF + (((j >> 1) * CAT_DIM + o) << 1) + (j & 1)] = s;
  }
}

// ---------------------------------------------------------------------------
// Fused per-edge kernel. One wave32 owns a tile of 16 edges.
// D = e_rbf_tile[16x16] @ M[16x272] via v_wmma_f32_16x16x4_f32,
// epilogue adds Ta'[z0]+Tb[z1] from LDS, swish (exp + v_rcp), NT store.
// ---------------------------------------------------------------------------
__global__ void emb_fused_kernel(const float* __restrict__ e_rbf,
                                 const int*   __restrict__ z,
                                 const int*   __restrict__ nbr,
                                 const float* __restrict__ ws,
                                 float*       __restrict__ out,
                                 int ntiles) {
  extern __shared__ float lds[];
  for (int i = threadIdx.x; i < PRE_TOT; i += BLOCK) lds[i] = ws[i];
  __syncthreads();

  const int lane = threadIdx.x & 31;
  const int wave = threadIdx.x >> 5;
  const int half = lane >> 4;   // 0: lanes 0-15, 1: lanes 16-31
  const int lrow = lane & 15;

  const float* sM2 = lds + M2_OFF;

  for (int tile = blockIdx.x * WAVES + wave; tile < ntiles;
       tile += gridDim.x * WAVES) {
    const int ebase = tile * 16;

    // A fragments: 16x4 f32 layout -> lane holds row (lrow), K = 4*kk + 2*half + {0,1}
    v2f a[4];
    const float* er = e_rbf + (size_t)(ebase + lrow) * N_RBF + 2 * half;
#pragma unroll
    for (int kk = 0; kk < 4; ++kk)
      a[kk] = *(const v2f*)(er + 4 * kk);

    // per-edge table row offsets (lane e<16 loads edge e; broadcast via shfl)
    const int et = ebase + lrow;
    const int n0 = nbr[2 * et];
    const int n1 = nbr[2 * et + 1];
    const int za = z[n0] * CAT_DIM;
    const int zb = z[n1] * CAT_DIM;
    int zaoff[8], zboff[8];
#pragma unroll
    for (int r = 0; r < 8; ++r) {
      zaoff[r] = __shfl(za, r + 8 * half, 32);
      zboff[r] = __shfl(zb, r + 8 * half, 32);
    }

#pragma unroll 1
    for (int o = 0; o < CAT_DIM / 16; ++o) {          // 17 output blocks
      const int col = o * 16 + lrow;
      v8f acc = {};
#pragma unroll
      for (int kk = 0; kk < 4; ++kk) {
        // B fragment: lane holds N=lrow, K rows (4*kk+2*half, +1) as float2
        v2f b = *(const v2f*)(sM2 + (((2 * kk + half) * CAT_DIM + col) << 1));
        acc = __builtin_amdgcn_wmma_f32_16x16x4_f32(
            false, a[kk], false, b, (short)0, acc, false, false);
      }
#pragma unroll
      for (int r = 0; r < 8; ++r) {
        // D layout: VGPR r, this half -> edge = ebase + r + 8*half, N = col
        float add = lds[TA_OFF + zaoff[r] + col] +
                    lds[TB_OFF + zboff[r] + col];
        float x = acc[r] + add;
        // swish: x * sigmoid(x); fast v_rcp instead of IEEE divide
        float y = x * __builtin_amdgcn_rcpf(1.0f + __expf(-x));
        __builtin_nontemporal_store(
            y, out + (size_t)(ebase + r + 8 * half) * CAT_DIM + col);
      }
    }
  }
}

// ---------------------------------------------------------------------------
extern "C" void kernel_launch(void* const* d_in, const int* in_sizes, int n_in,
                              void* d_out, int out_size, void* d_ws, size_t ws_size,
                              hipStream_t stream) {
  const float* e_rbf  = (const float*)d_in[0];
  const int*   zp     = (const int*)  d_in[1];
  const int*   nbr    = (const int*)  d_in[2];
  const float* edge_W = (const float*)d_in[3];
  const float* emb    = (const float*)d_in[4];
  const float* dW     = (const float*)d_in[5];
  const float* db     = (const float*)d_in[6];
  float* out = (float*)d_out;
  float* ws  = (float*)d_ws;

  const int E = in_sizes[0] / N_RBF;      // 800000
  const int ntiles = E / 16;              // 50000 (E is a multiple of 16)

  emb_precompute_kernel<<<(PRE_TOT + 255) / 256, 256, 0, stream>>>(
      edge_W, emb, dW, db, ws);

  const size_t smem = (size_t)LDS_FLOATS * sizeof(float);  // ~235 KB < 320 KB/WGP
  hipFuncSetAttribute((const void*)emb_fused_kernel,
                      hipFuncAttributeMaxDynamicSharedMemorySize, (int)smem);

  int blocks = 512;
  if (blocks > (ntiles + WAVES - 1) / WAVES) blocks = (ntiles + WAVES - 1) / WAVES;
  emb_fused_kernel<<<blocks, BLOCK, smem, stream>>>(
      e_rbf, zp, nbr, ws, out, ntiles);
}